// ENIGMAEncoder_45569603010801
// MI455X (gfx1250) — compile-verified
//
#include <hip/hip_runtime.h>
#include <math.h>

// ---- problem constants (from reference) ----
#define BB   32
#define CC   122
#define TT   500     // time / contraction length of subject layer
#define WID  80
#define UU   96      // pooled time length
#define K2   (WID*CC)   // 9760  conv2 contraction
#define HID  768
#define PROJ 1024
#define PLEN 484     // pooled-prefix length: m in [0, 483], P[m] = mean(in[m..m+16])
#define EPSF 1e-5f

typedef float v2f __attribute__((ext_vector_type(2)));
typedef float v8f __attribute__((ext_vector_type(8)));

// f32 WMMA 16x16x4:  D = A(16x4) * B(4x16) + C(16x16)
// A: lane = M (lane&15), half-wave (lane>>4) selects K pair {0,1}/{2,3}; VGPR0/1 = K within pair
// B: lane = N (lane&15), same half-wave K split  -> both operands read CONTIGUOUS K per lane
static __device__ __forceinline__ v8f wmma_f32(v2f a, v2f b, v8f c) {
    return __builtin_amdgcn_wmma_f32_16x16x4_f32(false, a, false, b, (short)0, c, false, false);
}

// ============================================================================
// K1: subject-specific linear layer.  out0[b,c,t] = sum_s x[b,c,s]*Wg[b,t,s] + bg[b,t]
// grid (8 Mtiles, 32 Ntiles, 32 batches), 1 wave per block.
// ============================================================================
__global__ __launch_bounds__(32) void k_subject(
    const float* __restrict__ x, const int* __restrict__ sid,
    const float* __restrict__ W_sub, const float* __restrict__ b_sub,
    float* __restrict__ out0)
{
    const int lane = threadIdx.x;
    const int b  = blockIdx.z;
    const int m0 = blockIdx.x * 16;          // over C=122
    const int n0 = blockIdx.y * 16;          // over T=500
    const int s  = sid[b];
    const float* Wg = W_sub + (size_t)s * TT * TT;

    int am = m0 + (lane & 15); if (am >= CC) am = CC - 1;   // clamp, discard at store
    int an = n0 + (lane & 15); if (an >= TT) an = TT - 1;
    const int koff = (lane >> 4) * 2;
    const float* pa = x  + ((size_t)b * CC + am) * TT + koff;   // contiguous in s
    const float* pb = Wg + (size_t)an * TT + koff;              // contiguous in s

    v8f acc = {};
    #pragma unroll 4
    for (int k4 = 0; k4 < TT / 4; ++k4) {
        v2f a = *(const v2f*)(pa + 4 * k4);
        v2f w = *(const v2f*)(pb + 4 * k4);
        acc = wmma_f32(a, w, acc);
    }

    const float* bg = b_sub + (size_t)s * TT;
    const int nn = n0 + (lane & 15);
    #pragma unroll
    for (int r = 0; r < 8; ++r) {
        int mm = m0 + r + ((lane >> 4) << 3);
        if (mm < CC && nn < TT)
            out0[((size_t)b * CC + mm) * TT + nn] = acc[r] + bg[nn];
    }
}

// ============================================================================
// K2: pre-pooled signal  P[b,c,m] = (1/17) * sum_{i=0..16} out0[b,c,m+i]
//     (avgpool(17,5) o conv1(1x5)  ==  5-tap conv on P sampled at stride 5)
// ============================================================================
__global__ __launch_bounds__(128) void k_pool(
    const float* __restrict__ out0, float* __restrict__ P)
{
    __shared__ float row[TT];
    const int bc = blockIdx.x;                    // b*C + c
    const float* src = out0 + (size_t)bc * TT;
    for (int i = threadIdx.x; i < TT; i += 128) row[i] = src[i];
    __syncthreads();
    float* dst = P + (size_t)bc * PLEN;
    for (int m = threadIdx.x; m < PLEN; m += 128) {
        float sum = 0.f;
        #pragma unroll
        for (int i = 0; i < 17; ++i) sum += row[m + i];
        dst[m] = sum * (1.f / 17.f);
    }
}

// ============================================================================
// K3: conv2 as WMMA GEMM, B-matrix generated on the fly (never materialize h1):
//     out2[b,w2,u] = sum_{k=(w,c)} conv2_w[w2][k] * elu(bn1(sum_j w1[w,j]*P[b,c,5u+j]))
// then +conv2_b, bn2, elu at store.
// grid (6 u-tiles, 32 batches), 160 threads = 5 waves (one M-tile of 16 each; 5*16 = 80 = WID).
// ============================================================================
__global__ __launch_bounds__(160) void k_conv2(
    const float* __restrict__ P,
    const float* __restrict__ conv1_w, const float* __restrict__ conv1_b,
    const float* __restrict__ bn1_g, const float* __restrict__ bn1_b,
    const float* __restrict__ bn1_m, const float* __restrict__ bn1_v,
    const float* __restrict__ conv2_w, const float* __restrict__ conv2_b,
    const float* __restrict__ bn2_g, const float* __restrict__ bn2_b,
    const float* __restrict__ bn2_m, const float* __restrict__ bn2_v,
    float* __restrict__ h2)
{
    __shared__ float Pl[CC * 80];      // P[b, c, 5*u0 + q], q in [0,80)  (39 KB)
    __shared__ float w1s[WID * 5];
    __shared__ float sc1[WID], sh1[WID];

    const int tid = threadIdx.x;
    const int b   = blockIdx.y;
    const int u0  = blockIdx.x * 16;

    for (int i = tid; i < WID * 5; i += 160) w1s[i] = conv1_w[i];
    if (tid < WID) {
        float sc = bn1_g[tid] * rsqrtf(bn1_v[tid] + EPSF);
        sc1[tid] = sc;
        sh1[tid] = bn1_b[tid] - bn1_m[tid] * sc + sc * conv1_b[tid];  // fold conv1 bias
    }
    for (int i = tid; i < CC * 80; i += 160) {
        int c = i / 80, q = i - c * 80;
        Pl[i] = P[((size_t)b * CC + c) * PLEN + 5 * u0 + q];
    }
    __syncthreads();

    const int wave = tid >> 5;            // M-tile: w2 in [wave*16, wave*16+16)
    const int lane = tid & 31;
    const int n    = lane & 15;           // N = u - u0
    const int half = lane >> 4;

    const float* paA = conv2_w + (size_t)(wave * 16 + (lane & 15)) * K2 + half * 2;

    int w0 = 0, c0 = half * 2;            // incremental k -> (w,c) decomposition, step 4
    v8f acc = {};
    for (int k4 = 0; k4 < K2 / 4; ++k4) {
        v2f a = *(const v2f*)(paA + 4 * k4);

        // B element 0: k = 4*k4 + 2*half  -> (w0, c0)
        const float* wr0 = &w1s[w0 * 5];
        const float* pr0 = &Pl[c0 * 80 + 5 * n];
        float v0 = wr0[0]*pr0[0] + wr0[1]*pr0[1] + wr0[2]*pr0[2] + wr0[3]*pr0[3] + wr0[4]*pr0[4];
        float g0 = sc1[w0] * v0 + sh1[w0];
        float e0 = g0 > 0.f ? g0 : (__expf(g0) - 1.f);

        // B element 1: k+1
        int c1 = c0 + 1, w1i = w0;
        if (c1 == CC) { c1 = 0; w1i = w0 + 1; }
        const float* wr1 = &w1s[w1i * 5];
        const float* pr1 = &Pl[c1 * 80 + 5 * n];
        float v1 = wr1[0]*pr1[0] + wr1[1]*pr1[1] + wr1[2]*pr1[2] + wr1[3]*pr1[3] + wr1[4]*pr1[4];
        float g1 = sc1[w1i] * v1 + sh1[w1i];
        float e1 = g1 > 0.f ? g1 : (__expf(g1) - 1.f);

        v2f bb = { e0, e1 };
        acc = wmma_f32(a, bb, acc);

        c0 += 4;
        if (c0 >= CC) { c0 -= CC; w0 += 1; }
    }

    #pragma unroll
    for (int r = 0; r < 8; ++r) {
        int w2 = wave * 16 + r + half * 8;            // always < 80
        float sc = bn2_g[w2] * rsqrtf(bn2_v[w2] + EPSF);
        float sh = bn2_b[w2] - bn2_m[w2] * sc;
        float v  = sc * (acc[r] + conv2_b[w2]) + sh;
        float e  = v > 0.f ? v : (__expf(v) - 1.f);
        h2[((size_t)b * WID + w2) * UU + u0 + n] = e;
    }
}

// ============================================================================
// K4: 1x1 proj (8 x 80) + transpose/reshape into z[b, u*8+e]
// ============================================================================
__global__ __launch_bounds__(256) void k_proj(
    const float* __restrict__ h2, const float* __restrict__ proj_w,
    const float* __restrict__ proj_b, float* __restrict__ z)
{
    __shared__ float hs[WID * UU];     // 30 KB
    __shared__ float pw[8 * WID];
    const int b = blockIdx.x;
    for (int i = threadIdx.x; i < WID * UU; i += 256) hs[i] = h2[(size_t)b * WID * UU + i];
    for (int i = threadIdx.x; i < 8 * WID; i += 256)  pw[i] = proj_w[i];
    __syncthreads();
    for (int idx = threadIdx.x; idx < HID; idx += 256) {
        int u = idx >> 3, e = idx & 7;
        float s = proj_b[e];
        #pragma unroll 8
        for (int w = 0; w < WID; ++w) s += pw[e * WID + w] * hs[w * UU + u];
        z[(size_t)b * HID + idx] = s;
    }
}

// ============================================================================
// K5/K6: MLP layers as WMMA GEMMs.  M=32 rows(batch), N=1024, contiguous K.
// ============================================================================
__global__ __launch_bounds__(32) void k_mlp1(
    const float* __restrict__ z, const float* __restrict__ w,
    const float* __restrict__ bias, float* __restrict__ z1, float* __restrict__ g)
{
    const int lane = threadIdx.x;
    const int m0 = blockIdx.x * 16, n0 = blockIdx.y * 16;
    const float* pa = z + (size_t)(m0 + (lane & 15)) * HID + (lane >> 4) * 2;
    const float* pb = w + (size_t)(n0 + (lane & 15)) * HID + (lane >> 4) * 2;
    v8f acc = {};
    #pragma unroll 4
    for (int k4 = 0; k4 < HID / 4; ++k4) {
        v2f a = *(const v2f*)(pa + 4 * k4);
        v2f bb = *(const v2f*)(pb + 4 * k4);
        acc = wmma_f32(a, bb, acc);
    }
    const int nn = n0 + (lane & 15);
    const float bn = bias[nn];
    #pragma unroll
    for (int r = 0; r < 8; ++r) {
        int mm = m0 + r + ((lane >> 4) << 3);
        float v = acc[r] + bn;
        z1[(size_t)mm * PROJ + nn] = v;
        g [(size_t)mm * PROJ + nn] = 0.5f * v * (1.f + erff(v * 0.70710678118654752f)); // exact gelu
    }
}

__global__ __launch_bounds__(32) void k_mlp2(
    const float* __restrict__ g, const float* __restrict__ w,
    const float* __restrict__ bias, const float* __restrict__ z1,
    float* __restrict__ y)
{
    const int lane = threadIdx.x;
    const int m0 = blockIdx.x * 16, n0 = blockIdx.y * 16;
    const float* pa = g + (size_t)(m0 + (lane & 15)) * PROJ + (lane >> 4) * 2;
    const float* pb = w + (size_t)(n0 + (lane & 15)) * PROJ + (lane >> 4) * 2;
    v8f acc = {};
    #pragma unroll 4
    for (int k4 = 0; k4 < PROJ / 4; ++k4) {
        v2f a = *(const v2f*)(pa + 4 * k4);
        v2f bb = *(const v2f*)(pb + 4 * k4);
        acc = wmma_f32(a, bb, acc);
    }
    const int nn = n0 + (lane & 15);
    const float bn = bias[nn];
    #pragma unroll
    for (int r = 0; r < 8; ++r) {
        int mm = m0 + r + ((lane >> 4) << 3);
        y[(size_t)mm * PROJ + nn] = z1[(size_t)mm * PROJ + nn] + acc[r] + bn;  // residual
    }
}

// ============================================================================
// K7: LayerNorm over last dim (1024) per batch row.
// ============================================================================
__global__ __launch_bounds__(256) void k_ln(
    const float* __restrict__ y, const float* __restrict__ gam,
    const float* __restrict__ bet, float* __restrict__ out)
{
    __shared__ float red[256];
    const int b = blockIdx.x;
    const float* row = y + (size_t)b * PROJ;
    float s = 0.f;
    for (int i = threadIdx.x; i < PROJ; i += 256) s += row[i];
    red[threadIdx.x] = s; __syncthreads();
    for (int off = 128; off > 0; off >>= 1) {
        if (threadIdx.x < off) red[threadIdx.x] += red[threadIdx.x + off];
        __syncthreads();
    }
    const float mu = red[0] * (1.f / PROJ);
    __syncthreads();
    float s2 = 0.f;
    for (int i = threadIdx.x; i < PROJ; i += 256) { float d = row[i] - mu; s2 += d * d; }
    red[threadIdx.x] = s2; __syncthreads();
    for (int off = 128; off > 0; off >>= 1) {
        if (threadIdx.x < off) red[threadIdx.x] += red[threadIdx.x + off];
        __syncthreads();
    }
    const float inv = rsqrtf(red[0] * (1.f / PROJ) + EPSF);
    for (int i = threadIdx.x; i < PROJ; i += 256)
        out[(size_t)b * PROJ + i] = (row[i] - mu) * inv * gam[i] + bet[i];
}

// ============================================================================
extern "C" void kernel_launch(void* const* d_in, const int* in_sizes, int n_in,
                              void* d_out, int out_size, void* d_ws, size_t ws_size,
                              hipStream_t stream) {
    const float* x       = (const float*)d_in[0];
    const int*   sid     = (const int*)  d_in[1];
    const float* W_sub   = (const float*)d_in[2];
    const float* b_sub   = (const float*)d_in[3];
    const float* conv1_w = (const float*)d_in[4];
    const float* conv1_b = (const float*)d_in[5];
    const float* bn1_g   = (const float*)d_in[6];
    const float* bn1_b   = (const float*)d_in[7];
    const float* bn1_m   = (const float*)d_in[8];
    const float* bn1_v   = (const float*)d_in[9];
    const float* conv2_w = (const float*)d_in[10];
    const float* conv2_b = (const float*)d_in[11];
    const float* bn2_g   = (const float*)d_in[12];
    const float* bn2_b   = (const float*)d_in[13];
    const float* bn2_m   = (const float*)d_in[14];
    const float* bn2_v   = (const float*)d_in[15];
    const float* proj_w  = (const float*)d_in[16];
    const float* proj_b  = (const float*)d_in[17];
    const float* mlp1_w  = (const float*)d_in[18];
    const float* mlp1_b  = (const float*)d_in[19];
    const float* mlp2_w  = (const float*)d_in[20];
    const float* mlp2_b  = (const float*)d_in[21];
    const float* ln_g    = (const float*)d_in[22];
    const float* ln_b    = (const float*)d_in[23];

    // workspace layout (floats), ~17 MB total
    float* ws   = (float*)d_ws;
    float* out0 = ws;                                   // B*C*T      = 1,952,000
    float* P    = out0 + (size_t)BB * CC * TT;          // B*C*484    = 1,889,536
    float* h2   = P    + (size_t)BB * CC * PLEN;        // B*80*96    =   245,760
    float* z    = h2   + (size_t)BB * WID * UU;         // B*768
    float* z1   = z    + (size_t)BB * HID;              // B*1024
    float* gg   = z1   + (size_t)BB * PROJ;             // B*1024
    float* yy   = gg   + (size_t)BB * PROJ;             // B*1024

    dim3 g1(8, 32, BB);                                 // Mtiles(C) x Ntiles(T) x batch
    k_subject<<<g1, 32, 0, stream>>>(x, sid, W_sub, b_sub, out0);

    k_pool<<<BB * CC, 128, 0, stream>>>(out0, P);

    dim3 g3(UU / 16, BB);
    k_conv2<<<g3, 160, 0, stream>>>(P, conv1_w, conv1_b, bn1_g, bn1_b, bn1_m, bn1_v,
                                    conv2_w, conv2_b, bn2_g, bn2_b, bn2_m, bn2_v, h2);

    k_proj<<<BB, 256, 0, stream>>>(h2, proj_w, proj_b, z);

    dim3 g5(2, PROJ / 16);
    k_mlp1<<<g5, 32, 0, stream>>>(z, mlp1_w, mlp1_b, z1, gg);
    k_mlp2<<<g5, 32, 0, stream>>>(gg, mlp2_w, mlp2_b, z1, yy);

    k_ln<<<BB, 256, 0, stream>>>(yy, ln_g, ln_b, (float*)d_out);
}